// SymEig_7181185319633
// MI455X (gfx1250) — compile-verified
//
#include <hip/hip_runtime.h>
#include <hip/hip_bf16.h>
#include <math.h>

// Batched 3x3 symmetric eigendecomposition, HBM-bandwidth-bound streaming
// kernel for MI455X (gfx1250, wave32). ~176 MB total traffic -> ~7.6us floor
// at 23.3 TB/s. No GEMM structure -> WMMA inapplicable; we instead exercise
// the CDNA5 async global->LDS path + wide B128 LDS-staged I/O.

#ifndef __has_builtin
#define __has_builtin(x) 0
#endif

#if __has_builtin(__builtin_amdgcn_global_load_async_to_lds_b128) && \
    __has_builtin(__builtin_amdgcn_s_wait_asynccnt)
#define HAVE_ASYNC_LOAD_LDS 1
#else
#define HAVE_ASYNC_LOAD_LDS 0
#endif

#if __has_builtin(__builtin_amdgcn_global_store_async_from_lds_b128) && \
    __has_builtin(__builtin_amdgcn_s_wait_asynccnt)
#define HAVE_ASYNC_STORE_LDS 1
#else
#define HAVE_ASYNC_STORE_LDS 0
#endif

#define TPB 256  // 8 waves (wave32) per block; 256 matrices per block

// Clang-native 16B vector for staged copies (works with nontemporal builtins
// on both host and device passes, unlike HIP_vector_type float4).
typedef int v4i __attribute__((ext_vector_type(4)));
// Async builtin signature (from hipcc diagnostic): (v4i AS1*, v4i AS3*, Ii, Ii)
typedef __attribute__((address_space(1))) v4i gv4i_t;
typedef __attribute__((address_space(3))) v4i lv4i_t;

__device__ __forceinline__ void cross3(float ax, float ay, float az,
                                       float bx, float by, float bz,
                                       float& cx, float& cy, float& cz) {
  cx = ay * bz - az * by;
  cy = az * bx - ax * bz;
  cz = ax * by - ay * bx;
}

// Eigenvector of symmetric A for eigenvalue lam via best cross product of
// rows of (A - lam*I). Returns false if all candidates are ~null (degenerate).
__device__ __forceinline__ bool eigvec3(float a00, float a01, float a02,
                                        float a11, float a12, float a22,
                                        float lam,
                                        float& vx, float& vy, float& vz) {
  float r0x = a00 - lam, r0y = a01,       r0z = a02;
  float r1x = a01,       r1y = a11 - lam, r1z = a12;
  float r2x = a02,       r2y = a12,       r2z = a22 - lam;
  float cax, cay, caz, cbx, cby, cbz, ccx, ccy, ccz;
  cross3(r0x, r0y, r0z, r1x, r1y, r1z, cax, cay, caz);
  cross3(r0x, r0y, r0z, r2x, r2y, r2z, cbx, cby, cbz);
  cross3(r1x, r1y, r1z, r2x, r2y, r2z, ccx, ccy, ccz);
  float na = cax * cax + cay * cay + caz * caz;
  float nb = cbx * cbx + cby * cby + cbz * cbz;
  float nc = ccx * ccx + ccy * ccy + ccz * ccz;
  float bx = cax, by = cay, bz = caz, bn = na;
  if (nb > bn) { bx = cbx; by = cby; bz = cbz; bn = nb; }
  if (nc > bn) { bx = ccx; by = ccy; bz = ccz; bn = nc; }
  bool ok = bn > 1e-28f;
  float inv = rsqrtf(fmaxf(bn, 1e-36f));
  vx = bx * inv; vy = by * inv; vz = bz * inv;
  return ok;
}

// Closed-form symmetric 3x3 eigensolve. w ascending (matches jnp.linalg.eigh),
// V row-major with eigenvectors in COLUMNS: V[r*3+c] = c-th eigvec, row r.
__device__ __forceinline__ void symeig3_core(float a00, float a01, float a02,
                                             float a11, float a12, float a22,
                                             float w[3], float V[9]) {
  const float THIRD = 1.0f / 3.0f;
  float q   = (a00 + a11 + a22) * THIRD;
  float b00 = a00 - q, b11 = a11 - q, b22 = a22 - q;
  float p1  = a01 * a01 + a02 * a02 + a12 * a12;
  float p2  = b00 * b00 + b11 * b11 + b22 * b22 + 2.0f * p1;
  float p   = sqrtf(p2 * (1.0f / 6.0f));
  float invp = (p > 1e-20f) ? (1.0f / p) : 0.0f;
  float c00 = b00 * invp, c11 = b11 * invp, c22 = b22 * invp;
  float d01 = a01 * invp, d02 = a02 * invp, d12 = a12 * invp;
  float detB = c00 * (c11 * c22 - d12 * d12)
             - d01 * (d01 * c22 - d12 * d02)
             + d02 * (d01 * d12 - c11 * d02);
  float r    = fminf(fmaxf(0.5f * detB, -1.0f), 1.0f);
  float phi  = acosf(r) * THIRD;
  float twop = 2.0f * p;
  float ehi  = q + twop * cosf(phi);                        // largest
  float elo  = q + twop * cosf(phi + 2.0943951023931953f);  // smallest (+2pi/3)
  float emid = 3.0f * q - ehi - elo;
  w[0] = elo; w[1] = emid; w[2] = ehi;

  // Extreme eigenvectors via cross products; middle one closes the frame.
  float v2x, v2y, v2z;
  bool ok2 = eigvec3(a00, a01, a02, a11, a12, a22, ehi, v2x, v2y, v2z);
  if (!ok2) { v2x = 1.0f; v2y = 0.0f; v2z = 0.0f; }  // scalar matrix
  float v0x, v0y, v0z;
  bool ok0 = eigvec3(a00, a01, a02, a11, a12, a22, elo, v0x, v0y, v0z);
  float dt = v0x * v2x + v0y * v2y + v0z * v2z;  // Gram-Schmidt vs v2
  v0x -= dt * v2x; v0y -= dt * v2y; v0z -= dt * v2z;
  float n0 = v0x * v0x + v0y * v0y + v0z * v0z;
  if (!ok0 || n0 < 1e-12f) {
    // degenerate pair: any unit vector orthogonal to v2
    float ax = fabsf(v2x), ay = fabsf(v2y), az = fabsf(v2z);
    float ux, uy, uz;
    if (ax <= ay && ax <= az) { ux = 1.0f; uy = 0.0f; uz = 0.0f; }
    else if (ay <= az)        { ux = 0.0f; uy = 1.0f; uz = 0.0f; }
    else                      { ux = 0.0f; uy = 0.0f; uz = 1.0f; }
    float dd = ux * v2x + uy * v2y + uz * v2z;
    v0x = ux - dd * v2x; v0y = uy - dd * v2y; v0z = uz - dd * v2z;
    n0 = v0x * v0x + v0y * v0y + v0z * v0z;
  }
  float inv0 = rsqrtf(fmaxf(n0, 1e-36f));
  v0x *= inv0; v0y *= inv0; v0z *= inv0;
  float v1x, v1y, v1z;
  cross3(v2x, v2y, v2z, v0x, v0y, v0z, v1x, v1y, v1z);
  float n1   = v1x * v1x + v1y * v1y + v1z * v1z;
  float inv1 = rsqrtf(fmaxf(n1, 1e-36f));
  v1x *= inv1; v1y *= inv1; v1z *= inv1;

  V[0] = v0x; V[1] = v1x; V[2] = v2x;
  V[3] = v0y; V[4] = v1y; V[5] = v2y;
  V[6] = v0z; V[7] = v1z; V[8] = v2z;
}

// Main kernel: 256 matrices per block, LDS-staged B128 coalesced I/O.
__global__ __launch_bounds__(TPB) void symeig3_main(
    const float* __restrict__ in, float* __restrict__ evals,
    float* __restrict__ evecs) {
  __shared__ __align__(16) float lin[TPB * 9];    //  9216 B in-staging
  __shared__ __align__(16) float lval[TPB * 3];   //  3072 B eigenvalues
  __shared__ __align__(16) float lvec[TPB * 9];   //  9216 B eigenvectors

  const int t = threadIdx.x;
  const size_t base = (size_t)blockIdx.x * TPB;

  // ---- cooperative load: 576 x b128 (16B-aligned: base*9*4 = 9216B*blk) ---
  const v4i* gsrc = (const v4i*)(in + base * 9);
  v4i* lin4 = (v4i*)lin;
#if HAVE_ASYNC_LOAD_LDS
  {
    __builtin_amdgcn_global_load_async_to_lds_b128(
        (gv4i_t*)(gsrc + t),       (lv4i_t*)(lin4 + t),       0, 0);
    __builtin_amdgcn_global_load_async_to_lds_b128(
        (gv4i_t*)(gsrc + t + 256), (lv4i_t*)(lin4 + t + 256), 0, 0);
    if (t < 64)
      __builtin_amdgcn_global_load_async_to_lds_b128(
          (gv4i_t*)(gsrc + t + 512), (lv4i_t*)(lin4 + t + 512), 0, 0);
    __builtin_amdgcn_s_wait_asynccnt(0);
  }
#else
  {
    lin4[t]       = __builtin_nontemporal_load(gsrc + t);
    lin4[t + 256] = __builtin_nontemporal_load(gsrc + t + 256);
    if (t < 64) lin4[t + 512] = __builtin_nontemporal_load(gsrc + t + 512);
  }
#endif
  __syncthreads();

  // ---- per-thread solve (stride-9 dword LDS reads: bank-conflict-free) ----
  {
    const float* m = &lin[t * 9];
    float a00 = m[0], a11 = m[4], a22 = m[8];
    float a01 = 0.5f * (m[1] + m[3]);
    float a02 = 0.5f * (m[2] + m[6]);
    float a12 = 0.5f * (m[5] + m[7]);
    float w[3], V[9];
    symeig3_core(a00, a01, a02, a11, a12, a22, w, V);
    lval[t * 3 + 0] = w[0];
    lval[t * 3 + 1] = w[1];
    lval[t * 3 + 2] = w[2];
    float* o = &lvec[t * 9];
#pragma unroll
    for (int k = 0; k < 9; ++k) o[k] = V[k];
  }
  __syncthreads();

  // ---- cooperative store: 192 + 576 x b128 -------------------------------
  v4i* gval = (v4i*)(evals + base * 3);
  v4i* gvec = (v4i*)(evecs + base * 9);
  const v4i* lv4 = (const v4i*)lval;
  const v4i* le4 = (const v4i*)lvec;
#if HAVE_ASYNC_STORE_LDS
  if (t < 192)
    __builtin_amdgcn_global_store_async_from_lds_b128(
        (gv4i_t*)(gval + t), (lv4i_t*)(lv4 + t), 0, 0);
  __builtin_amdgcn_global_store_async_from_lds_b128(
      (gv4i_t*)(gvec + t),       (lv4i_t*)(le4 + t),       0, 0);
  __builtin_amdgcn_global_store_async_from_lds_b128(
      (gv4i_t*)(gvec + t + 256), (lv4i_t*)(le4 + t + 256), 0, 0);
  if (t < 64)
    __builtin_amdgcn_global_store_async_from_lds_b128(
        (gv4i_t*)(gvec + t + 512), (lv4i_t*)(le4 + t + 512), 0, 0);
  // ASYNCcnt drained by implicit wait-idle at S_ENDPGM.
#else
  if (t < 192) __builtin_nontemporal_store(lv4[t], gval + t);
  __builtin_nontemporal_store(le4[t], gvec + t);
  __builtin_nontemporal_store(le4[t + 256], gvec + t + 256);
  if (t < 64) __builtin_nontemporal_store(le4[t + 512], gvec + t + 512);
#endif
}

// Scalar tail for nmat not divisible by 256 (unused for 128^3, kept for
// generality/determinism).
__global__ void symeig3_tail(const float* __restrict__ in,
                             float* __restrict__ evals,
                             float* __restrict__ evecs, int start, int nmat) {
  int i = start + blockIdx.x * blockDim.x + threadIdx.x;
  if (i >= nmat) return;
  const float* m = in + (size_t)i * 9;
  float a00 = m[0], a11 = m[4], a22 = m[8];
  float a01 = 0.5f * (m[1] + m[3]);
  float a02 = 0.5f * (m[2] + m[6]);
  float a12 = 0.5f * (m[5] + m[7]);
  float w[3], V[9];
  symeig3_core(a00, a01, a02, a11, a12, a22, w, V);
  float* ev = evals + (size_t)i * 3;
  float* eg = evecs + (size_t)i * 9;
#pragma unroll
  for (int k = 0; k < 3; ++k) ev[k] = w[k];
#pragma unroll
  for (int k = 0; k < 9; ++k) eg[k] = V[k];
}

extern "C" void kernel_launch(void* const* d_in, const int* in_sizes, int n_in,
                              void* d_out, int out_size, void* d_ws,
                              size_t ws_size, hipStream_t stream) {
  (void)n_in; (void)d_ws; (void)ws_size; (void)out_size;
  const float* x = (const float*)d_in[0];
  const int nmat = in_sizes[0] / 9;  // 128^3 = 2,097,152
  float* evals = (float*)d_out;                    // nmat*3 floats
  float* evecs = evals + (size_t)nmat * 3;         // nmat*9 floats

  const int nfull = nmat / TPB;                    // 8192 full blocks
  if (nfull > 0) symeig3_main<<<nfull, TPB, 0, stream>>>(x, evals, evecs);

  const int rem = nmat - nfull * TPB;
  if (rem > 0) {
    int blocks = (rem + TPB - 1) / TPB;
    symeig3_tail<<<blocks, TPB, 0, stream>>>(x, evals, evecs, nfull * TPB,
                                             nmat);
  }
}